// STSIM_M_69630009803062
// MI455X (gfx1250) — compile-verified
//
#include <hip/hip_runtime.h>
#include <math.h>

typedef __attribute__((ext_vector_type(2))) float v2f;
typedef __attribute__((ext_vector_type(8))) float v8f;

#define FEAT 82
#define KPAD 84
#define NG   256

// ---------------------------------------------------------------------------
// Kernel 0: zero the 6*256-float workspace (count, sx, sy, sxy, sxx, syy).
// ---------------------------------------------------------------------------
__global__ void zero_ws_kernel(float* __restrict__ ws) {
  ws[blockIdx.x * 256 + threadIdx.x] = 0.0f;
}

// ---------------------------------------------------------------------------
// Kernel 1: pred[t] = sqrt( sum_d W[d] * (X1[t,d]-X2[t,d])^2 )
// via V_WMMA_F32_16X16X4_F32. One wave owns a 16-row tile; B-matrix has W[k]
// broadcast across all 16 columns so every D column equals the dot product.
// ---------------------------------------------------------------------------
__global__ void pred_wmma_kernel(const float* __restrict__ X1,
                                 const float* __restrict__ X2,
                                 const float* __restrict__ W,
                                 float* __restrict__ pred,
                                 int T) {
  __shared__ float Wl[KPAD];
  const int tid = threadIdx.x;
  if (tid < KPAD) Wl[tid] = (tid < FEAT) ? W[tid] : 0.0f;
  __syncthreads();

  const int wid  = tid >> 5;
  const int lane = tid & 31;
  const int base = (blockIdx.x * 8 + wid) * 16;
  if (base >= T) return;  // wave-uniform exit (no WMMA issued with partial EXEC)

  const int m    = lane & 15;
  const int koff = (lane < 16) ? 0 : 2;  // A/B K-pair held by this half-wave
  int row = base + m;
  if (row > T - 1) row = T - 1;          // clamp duplicate rows for tail tile

  const long long maxPair = ((long long)T * FEAT - 2) / 2;  // last valid float2
  const long long rowBase = (long long)row * FEAT;

  v8f acc = {};
#pragma unroll
  for (int k0 = 0; k0 < KPAD; k0 += 4) {   // 21 WMMA steps, K padded to 84
    long long flat = (rowBase + k0 + koff) >> 1;  // float2 index (8B aligned)
    if (flat > maxPair) flat = maxPair;           // OOB tail * W=0 -> harmless
    v2f x1 = ((const v2f*)X1)[flat];
    v2f x2 = ((const v2f*)X2)[flat];
    v2f a, b;
    float d0 = x1.x - x2.x;
    float d1 = x1.y - x2.y;
    a.x = d0 * d0;
    a.y = d1 * d1;
    b.x = Wl[k0 + koff];       // B[k][n] = W[k] for all n (replicated columns)
    b.y = Wl[k0 + koff + 1];
    acc = __builtin_amdgcn_wmma_f32_16x16x4_f32(
        /*neg_a=*/false, a, /*neg_b=*/false, b,
        /*c_mod=*/(short)0, acc, /*reuse_a=*/false, /*reuse_b=*/false);
  }

  // C/D layout: VGPR r, lanes 0-15 -> M=r ; lanes 16-31 -> M=8+r.
  // All N columns identical, so lane 0 carries rows base..base+7,
  // lane 16 carries rows base+8..base+15.
  if (lane == 0 || lane == 16) {
    const int r0 = base + ((lane == 16) ? 8 : 0);
#pragma unroll
    for (int i = 0; i < 8; ++i) {
      const int r = r0 + i;
      if (r < T) pred[r] = sqrtf(acc[i]);
    }
  }
}

// ---------------------------------------------------------------------------
// Kernel 2: per-group raw moments via LDS float atomics, then one global
// atomicAdd per (group, stat) per block. ws layout: ws[stat*256 + group].
// ---------------------------------------------------------------------------
__global__ void seg_stats_kernel(const float* __restrict__ pred,
                                 const float* __restrict__ Y,
                                 const int* __restrict__ mask,
                                 float* __restrict__ ws,
                                 int T) {
  __shared__ float s[6][NG];
  const int tid = threadIdx.x;
  for (int i = tid; i < 6 * NG; i += blockDim.x) (&s[0][0])[i] = 0.0f;
  __syncthreads();

  for (int t = blockIdx.x * blockDim.x + tid; t < T;
       t += gridDim.x * blockDim.x) {
    const float x = pred[t];
    const float y = Y[t];
    const int g = mask[t] & (NG - 1);
    atomicAdd(&s[0][g], 1.0f);
    atomicAdd(&s[1][g], x);
    atomicAdd(&s[2][g], y);
    atomicAdd(&s[3][g], x * y);
    atomicAdd(&s[4][g], x * x);
    atomicAdd(&s[5][g], y * y);
  }
  __syncthreads();

  for (int g = tid; g < NG; g += blockDim.x) {
#pragma unroll
    for (int i = 0; i < 6; ++i) atomicAdd(&ws[i * NG + g], s[i][g]);
  }
}

// ---------------------------------------------------------------------------
// Kernel 3: per-group Pearson |rho| from raw moments, reduce, /256.
// ---------------------------------------------------------------------------
__global__ void finalize_kernel(const float* __restrict__ ws,
                                float* __restrict__ out) {
  __shared__ float red[NG];
  const int g = threadIdx.x;
  const float c   = ws[0 * NG + g];
  const float sx  = ws[1 * NG + g];
  const float sy  = ws[2 * NG + g];
  const float sxy = ws[3 * NG + g];
  const float sxx = ws[4 * NG + g];
  const float syy = ws[5 * NG + g];

  float v = 0.0f;
  if (c > 0.0f) {
    const float nom = sxy - sx * sy / c;
    const float vx  = sxx - sx * sx / c;
    const float vy  = syy - sy * sy / c;
    const float den = sqrtf(vx * vy);
    v = fabsf(nom / den);   // matches reference (NaN/Inf propagate identically)
  }
  red[g] = v;
  __syncthreads();
  for (int step = NG / 2; step > 0; step >>= 1) {
    if (g < step) red[g] += red[g + step];
    __syncthreads();
  }
  if (g == 0) out[0] = red[0] / (float)NG;
}

// ---------------------------------------------------------------------------
extern "C" void kernel_launch(void* const* d_in, const int* in_sizes, int n_in,
                              void* d_out, int out_size, void* d_ws,
                              size_t ws_size, hipStream_t stream) {
  const float* X1   = (const float*)d_in[0];
  const float* X2   = (const float*)d_in[1];
  const float* Y    = (const float*)d_in[2];
  const float* W    = (const float*)d_in[3];
  const int*   mask = (const int*)d_in[4];

  const int T = in_sizes[2];            // 1,000,000 rows
  float* out  = (float*)d_out;          // [0] = coeff, [1..T] = pred
  float* pred = out + 1;
  float* ws   = (float*)d_ws;           // 6*256 floats

  zero_ws_kernel<<<6, 256, 0, stream>>>(ws);

  const int pblocks = (T + 127) / 128;  // 8 waves/block * 16 rows/wave
  pred_wmma_kernel<<<pblocks, 256, 0, stream>>>(X1, X2, W, pred, T);

  int sblocks = (T + 255) / 256;
  if (sblocks > 1024) sblocks = 1024;
  seg_stats_kernel<<<sblocks, 256, 0, stream>>>(pred, Y, mask, ws, T);

  finalize_kernel<<<1, NG, 0, stream>>>(ws, out);
}